// SpectTempFilts_87076166959529
// MI455X (gfx1250) — compile-verified
//
#include <hip/hip_runtime.h>
#include <hip/hip_bf16.h>

// ---------------------------------------------------------------------------
// SpectTempFilts on MI455X (gfx1250, wave32).
//
//   out[f] = mean over 211x390 crop of clamp(Re(ifft2(fft2(pad(100*x)) *
//            H_f))^2, 1e-8)
//
// FFTs restructured as dense f32 GEMMs on V_WMMA_F32_16X16X4_F32:
//   X    = F422 @ xp @ F1170          (forward FFT2, one-time)
//   U+iV = (X .* H_f) @ B             (B = cropped inverse DFT, /1170)
//   R    = Ar @ U - Ai @ V            (A = cropped inverse DFT, /422)
//   out[f] = mean(max(R^2, 1e-8))     (crop folded into A/B shapes)
//
// HBM floor: H = 470 MB streamed once (~20 us @ 23.3 TB/s); everything else
// sits in the 192 MB L2.  Design decisions driven by the ISA:
//  * all workspace matrices zero-padded to tile multiples -> unconditional
//    loads (no exec-mask branching; round-1 fix)
//  * B-side operands stored n-major so each 4x16 fragment needs one
//    global_load_b64 per two k-rows instead of two strided b32 (round-2 fix);
//    U/V tiles stored n-major with 2x global_store_b128
//  * quad-N tiles per wave (16 WMMAs per k-step in stage C) to amortize the
//    A-side H stream across 64 output columns
//  * raw H addresses clamped (v_min) instead of guarded; OOB lanes multiply
//    the zero-padded X spectrum
// ---------------------------------------------------------------------------

typedef __attribute__((ext_vector_type(2))) float v2f;
typedef __attribute__((ext_vector_type(4))) float v4f;
typedef __attribute__((ext_vector_type(8))) float v8f;

#define NF_    211          // valid freq (crop rows)
#define NT_    390          // valid time (crop cols)
#define MP     422          // FFT size, freq
#define NP     1170         // FFT size, time
#define MPP    432          // MP  -> mult of 16
#define NPP    1184         // NP  -> mult of 16
#define NFP    224          // NF_ -> mult of 16
#define NTP    448          // NT_ -> mult of 64 (quad-N tiles)
#define NFILT  119
#define FILT_STRIDE ((size_t)MP * NP)   // 493740 floats (H's real layout)
#define FCHUNK 8
#define TM_D   (NFP / 16)   // 14 m-tiles, stage D
#define TN4    (NTP / 64)   // 7 quad-N tile groups over time
#define NTILED (TM_D * TN4) // 98 partials per filter

// ---- WMMA fragment helpers (ISA 7.12.2 layouts, wave32) --------------------

// A 16x4 f32 (row-major source): lane L = row M=L%16; VGPR v: K=k0+2*(L/16)+v.
__device__ __forceinline__ v2f load_a(const float* __restrict__ A, int ld,
                                      int m0, int k0) {
  int lane = threadIdx.x & 31;
  const float* p = A + (size_t)(m0 + (lane & 15)) * ld + (k0 + 2 * (lane >> 4));
  return *(const v2f*)p;                 // even index -> 8B aligned
}

// A fragment from raw H (ld = NP): clamp instead of guard (OOB lanes hit
// zero-padded X, contributing 0).
__device__ __forceinline__ v2f load_a_h(const float* __restrict__ H,
                                        int m0, int k0) {
  int lane = threadIdx.x & 31;
  int m = m0 + (lane & 15);
  int k = k0 + 2 * (lane >> 4);
  m = m < (MP - 1) ? m : (MP - 1);       // <= 421
  k = k < (NP - 2) ? k : (NP - 2);       // <= 1168 (even -> pair in range)
  return *(const v2f*)(H + (size_t)m * NP + k);
}

// B 4x16 f32 from an n-major (transposed) source Bt[n][k]:
// lane L = col N=L%16; VGPR v: K=k0+2*(L/16)+v -> adjacent -> one b64 load.
__device__ __forceinline__ v2f load_bt(const float* __restrict__ Bt, int ld,
                                       int k0, int n0) {
  int lane = threadIdx.x & 31;
  const float* p = Bt + (size_t)(n0 + (lane & 15)) * ld + (k0 + 2 * (lane >> 4));
  return *(const v2f*)p;
}

// C/D 16x16 f32, row-major store (8 b32 per lane; used only by the one-time FFT).
__device__ __forceinline__ void store_tile(float* __restrict__ D, int ld,
                                           int m0, int n0, v8f acc) {
  int lane = threadIdx.x & 31;
  int n = n0 + (lane & 15);
  int mb = m0 + 8 * (lane >> 4);
#pragma unroll
  for (int v = 0; v < 8; ++v) D[(size_t)(mb + v) * ld + n] = acc[v];
}

// C/D 16x16 f32, n-major store: 8 contiguous m-values -> 2x b128.
__device__ __forceinline__ void store_tile_t(float* __restrict__ Dt, int ld,
                                             int m0, int n0, v8f acc) {
  int lane = threadIdx.x & 31;
  int n = n0 + (lane & 15);
  int mb = m0 + 8 * (lane >> 4);              // mult of 8 -> 16B aligned
  float* p = Dt + (size_t)n * ld + mb;
  v4f lo = {acc[0], acc[1], acc[2], acc[3]};
  v4f hi = {acc[4], acc[5], acc[6], acc[7]};
  *(v4f*)p = lo;
  *(v4f*)(p + 4) = hi;
}

__device__ __forceinline__ v8f wmma4(v2f a, v2f b, v8f c) {
  return __builtin_amdgcn_wmma_f32_16x16x4_f32(
      false, a, false, b, (short)0, c, false, false);
}

// ---- setup kernels ---------------------------------------------------------

// Padded, rescaled input stored TRANSPOSED (time-major) as fwd1's B-operand.
__global__ void build_xp_kernel(const float* __restrict__ x,
                                float* __restrict__ xpt) {  // [NPP][MPP]
  int i = blockIdx.x * blockDim.x + threadIdx.x;   // freq [0,MPP)
  int j = blockIdx.y;                              // time [0,NPP)
  if (i >= MPP) return;
  float v = 0.f;
  if (i < NF_ && j < NT_) v = x[(size_t)i * NT_ + j] * 100.f;  // COCH_RESCALE
  xpt[(size_t)j * MPP + i] = v;
}

// Twiddles with explicit zero padding outside the valid region.
// Value depends only on r*c, so "transposed" generation is just swapped dims.
__global__ void fill_twiddle_kernel(float* __restrict__ R, float* __restrict__ I,
                                    int cols_alloc, int rows_valid,
                                    int cols_valid, int period,
                                    float sign, float scale) {
  int c = blockIdx.x * blockDim.x + threadIdx.x;
  int r = blockIdx.y;
  if (c >= cols_alloc) return;
  float cr = 0.f, ci = 0.f;
  if (r < rows_valid && c < cols_valid) {
    int t = (int)(((long long)r * c) % period);
    float ang = sign * 6.283185307179586f * (float)t / (float)period;
    float s, co;
    __sincosf(ang, &s, &co);
    cr = co * scale;
    ci = s * scale;
  }
  R[(size_t)r * cols_alloc + c] = cr;
  I[(size_t)r * cols_alloc + c] = ci;
}

// ---- forward FFT2 as GEMMs (one-time, dual-N per wave) ---------------------

// T = F422 @ xp.  M=MPP, K=MPP, N=NPP.  xp is n-major (xpt).
__global__ void gemm_fwd1_kernel(const float* __restrict__ Fr,
                                 const float* __restrict__ Fi,
                                 const float* __restrict__ xpt,
                                 float* __restrict__ Tr,
                                 float* __restrict__ Ti) {
  int n0 = blockIdx.x * 32;
  int m0 = blockIdx.y * 16;
  v8f ar0 = {}, ai0 = {}, ar1 = {}, ai1 = {};
#pragma unroll 2
  for (int k0 = 0; k0 < MPP; k0 += 4) {
    v2f fr = load_a(Fr, MPP, m0, k0);
    v2f fi = load_a(Fi, MPP, m0, k0);
    v2f x0 = load_bt(xpt, MPP, k0, n0);
    v2f x1 = load_bt(xpt, MPP, k0, n0 + 16);
    ar0 = wmma4(fr, x0, ar0);
    ai0 = wmma4(fi, x0, ai0);
    ar1 = wmma4(fr, x1, ar1);
    ai1 = wmma4(fi, x1, ai1);
  }
  store_tile(Tr, NPP, m0, n0, ar0);      // row-major: T is fwd2's A-operand
  store_tile(Ti, NPP, m0, n0, ai0);
  store_tile(Tr, NPP, m0, n0 + 16, ar1);
  store_tile(Ti, NPP, m0, n0 + 16, ai1);
}

// X = T @ G (complex x complex).  G is symmetric -> already its own transpose.
__global__ void gemm_fwd2_kernel(const float* __restrict__ Tr,
                                 const float* __restrict__ Ti,
                                 const float* __restrict__ Gr,
                                 const float* __restrict__ Gi,
                                 float* __restrict__ Xr,
                                 float* __restrict__ Xi) {
  int n0 = blockIdx.x * 32;
  int m0 = blockIdx.y * 16;
  v8f xr0 = {}, xi0 = {}, xr1 = {}, xi1 = {};
#pragma unroll 2
  for (int k0 = 0; k0 < NPP; k0 += 4) {
    v2f tr = load_a(Tr, NPP, m0, k0);
    v2f ti = load_a(Ti, NPP, m0, k0);
    v2f nti = -ti;
    v2f gr0 = load_bt(Gr, NPP, k0, n0);
    v2f gi0 = load_bt(Gi, NPP, k0, n0);
    v2f gr1 = load_bt(Gr, NPP, k0, n0 + 16);
    v2f gi1 = load_bt(Gi, NPP, k0, n0 + 16);
    xr0 = wmma4(tr, gr0, xr0);
    xr0 = wmma4(nti, gi0, xr0);
    xi0 = wmma4(tr, gi0, xi0);
    xi0 = wmma4(ti, gr0, xi0);
    xr1 = wmma4(tr, gr1, xr1);
    xr1 = wmma4(nti, gi1, xr1);
    xi1 = wmma4(tr, gi1, xi1);
    xi1 = wmma4(ti, gr1, xi1);
  }
  store_tile(Xr, NPP, m0, n0, xr0);      // row-major: X is stage C's A-operand
  store_tile(Xi, NPP, m0, n0, xi0);
  store_tile(Xr, NPP, m0, n0 + 16, xr1);
  store_tile(Xi, NPP, m0, n0 + 16, xi1);
}

// ---- stage C: U + iV = (X .* H_f) @ B.  M=MPP, K=NPP, N=NTP, quad-N --------
// Spectrum multiply fused into the A-operand: H streamed once from HBM.
// 16 WMMAs + 12 b64 loads per k-step; U/V written n-major via b128 stores.

__global__ void stageC_kernel(const float* __restrict__ Hr,   // + f0 offset
                              const float* __restrict__ Hi,
                              const float* __restrict__ Xr,
                              const float* __restrict__ Xi,
                              const float* __restrict__ Btr,   // [NTP][NPP]
                              const float* __restrict__ Bti,
                              float* __restrict__ Ut,          // [NTP][MPP]
                              float* __restrict__ Vt) {
  int z = blockIdx.z;
  const float* hr = Hr + (size_t)z * FILT_STRIDE;
  const float* hi = Hi + (size_t)z * FILT_STRIDE;
  float* ut = Ut + (size_t)z * NTP * MPP;
  float* vt = Vt + (size_t)z * NTP * MPP;
  int n0 = blockIdx.x * 64;
  int m0 = blockIdx.y * 16;
  v8f u0 = {}, v0 = {}, u1 = {}, v1 = {}, u2 = {}, v2 = {}, u3 = {}, v3 = {};
  for (int k0 = 0; k0 < NPP; k0 += 4) {
    // Keep the H stream ahead of the matrix pipes (global_prefetch_b8).
    __builtin_prefetch(hr + (size_t)m0 * NP + k0 + 256, 0, 1);
    __builtin_prefetch(hi + (size_t)m0 * NP + k0 + 256, 0, 1);
    v2f xr = load_a(Xr, NPP, m0, k0);
    v2f xi = load_a(Xi, NPP, m0, k0);
    v2f ahr = load_a_h(hr, m0, k0);
    v2f ahi = load_a_h(hi, m0, k0);
    v2f yr = xr * ahr - xi * ahi;        // Y = X .* H (VALU)
    v2f yi = xr * ahi + xi * ahr;
    v2f nyi = -yi;
    v2f br0 = load_bt(Btr, NPP, k0, n0);
    v2f bi0 = load_bt(Bti, NPP, k0, n0);
    v2f br1 = load_bt(Btr, NPP, k0, n0 + 16);
    v2f bi1 = load_bt(Bti, NPP, k0, n0 + 16);
    v2f br2 = load_bt(Btr, NPP, k0, n0 + 32);
    v2f bi2 = load_bt(Bti, NPP, k0, n0 + 32);
    v2f br3 = load_bt(Btr, NPP, k0, n0 + 48);
    v2f bi3 = load_bt(Bti, NPP, k0, n0 + 48);
    u0 = wmma4(yr, br0, u0); u0 = wmma4(nyi, bi0, u0);
    v0 = wmma4(yr, bi0, v0); v0 = wmma4(yi, br0, v0);
    u1 = wmma4(yr, br1, u1); u1 = wmma4(nyi, bi1, u1);
    v1 = wmma4(yr, bi1, v1); v1 = wmma4(yi, br1, v1);
    u2 = wmma4(yr, br2, u2); u2 = wmma4(nyi, bi2, u2);
    v2 = wmma4(yr, bi2, v2); v2 = wmma4(yi, br2, v2);
    u3 = wmma4(yr, br3, u3); u3 = wmma4(nyi, bi3, u3);
    v3 = wmma4(yr, bi3, v3); v3 = wmma4(yi, br3, v3);
  }
  store_tile_t(ut, MPP, m0, n0, u0);
  store_tile_t(vt, MPP, m0, n0, v0);
  store_tile_t(ut, MPP, m0, n0 + 16, u1);
  store_tile_t(vt, MPP, m0, n0 + 16, v1);
  store_tile_t(ut, MPP, m0, n0 + 32, u2);
  store_tile_t(vt, MPP, m0, n0 + 32, v2);
  store_tile_t(ut, MPP, m0, n0 + 48, u3);
  store_tile_t(vt, MPP, m0, n0 + 48, v3);
}

// ---- stage D: R = Ar@U - Ai@V, fused square/clamp/mean epilogue ------------
// M=NFP, K=MPP, N=NTP, quad-N.  U/V read n-major (b64 fragments).

__device__ __forceinline__ float tile_power(v8f acc, int m0, int n) {
  float s = 0.f;
  if (n < NT_) {
#pragma unroll
    for (int q = 0; q < 8; ++q) {
      if (m0 + q < NF_) {
        float r = acc[q];
        s += fmaxf(r * r, 1e-8f);        // SquareReal + clamp-min
      }
    }
  }
  return s;
}

__global__ void stageD_kernel(const float* __restrict__ Ut,
                              const float* __restrict__ Vt,
                              const float* __restrict__ Ar,
                              const float* __restrict__ Ai,
                              float* __restrict__ partial) {   // + f0*NTILED
  int z = blockIdx.z;
  const float* ut = Ut + (size_t)z * NTP * MPP;
  const float* vt = Vt + (size_t)z * NTP * MPP;
  int n0 = blockIdx.x * 64;
  int m0 = blockIdx.y * 16;
  v8f r0 = {}, r1 = {}, r2 = {}, r3 = {};
#pragma unroll 2
  for (int k0 = 0; k0 < MPP; k0 += 4) {
    v2f ar = load_a(Ar, MPP, m0, k0);
    v2f nai = -load_a(Ai, MPP, m0, k0);
    v2f ub0 = load_bt(ut, MPP, k0, n0);
    v2f vb0 = load_bt(vt, MPP, k0, n0);
    v2f ub1 = load_bt(ut, MPP, k0, n0 + 16);
    v2f vb1 = load_bt(vt, MPP, k0, n0 + 16);
    v2f ub2 = load_bt(ut, MPP, k0, n0 + 32);
    v2f vb2 = load_bt(vt, MPP, k0, n0 + 32);
    v2f ub3 = load_bt(ut, MPP, k0, n0 + 48);
    v2f vb3 = load_bt(vt, MPP, k0, n0 + 48);
    r0 = wmma4(ar, ub0, r0); r0 = wmma4(nai, vb0, r0);   // R = Ar*U - Ai*V
    r1 = wmma4(ar, ub1, r1); r1 = wmma4(nai, vb1, r1);
    r2 = wmma4(ar, ub2, r2); r2 = wmma4(nai, vb2, r2);
    r3 = wmma4(ar, ub3, r3); r3 = wmma4(nai, vb3, r3);
  }
  int lane = threadIdx.x & 31;
  int mb = m0 + 8 * (lane >> 4);
  int n = n0 + (lane & 15);
  float s = tile_power(r0, mb, n) + tile_power(r1, mb, n + 16) +
            tile_power(r2, mb, n + 32) + tile_power(r3, mb, n + 48);
#pragma unroll
  for (int off = 16; off > 0; off >>= 1) s += __shfl_xor(s, off, 32);
  if (lane == 0)
    partial[(size_t)z * NTILED + blockIdx.y * TN4 + blockIdx.x] = s;
}

// ---- stage E: per-filter reduction -> mean ---------------------------------

__global__ void reduce_out_kernel(const float* __restrict__ partial,
                                  float* __restrict__ out) {
  int f = blockIdx.x;
  __shared__ float sm[128];
  float s = 0.f;
  for (int i = threadIdx.x; i < NTILED; i += blockDim.x)
    s += partial[(size_t)f * NTILED + i];
  sm[threadIdx.x] = s;
  __syncthreads();
  for (int st = 64; st > 0; st >>= 1) {
    if ((int)threadIdx.x < st) sm[threadIdx.x] += sm[threadIdx.x + st];
    __syncthreads();
  }
  if (threadIdx.x == 0) out[f] = sm[0] * (1.f / (float)(NF_ * NT_));
}

// ---------------------------------------------------------------------------

extern "C" void kernel_launch(void* const* d_in, const int* in_sizes, int n_in,
                              void* d_out, int out_size, void* d_ws, size_t ws_size,
                              hipStream_t stream) {
  const float* x  = (const float*)d_in[0];   // [1,1,211,390]
  const float* Hr = (const float*)d_in[1];   // [119,422,1170]
  const float* Hi = (const float*)d_in[2];
  float* out = (float*)d_out;                // [119] f32
  float* ws = (float*)d_ws;

  // Workspace layout (floats); total ~10.1M floats = 40.4 MB.
  size_t o = 0;
  float* XPT  = ws + o; o += (size_t)NPP * MPP;     // padded input, n-major
  float* F4R  = ws + o; o += (size_t)MPP * MPP;     // fwd DFT 422 (padded)
  float* F4I  = ws + o; o += (size_t)MPP * MPP;
  float* GR   = ws + o; o += (size_t)NPP * NPP;     // fwd DFT 1170 (symmetric)
  float* GI   = ws + o; o += (size_t)NPP * NPP;
  float* TR   = ws + o; o += (size_t)MPP * NPP;     // T = F422 @ xp (row-major)
  float* TI   = ws + o; o += (size_t)MPP * NPP;
  float* XR   = ws + o; o += (size_t)MPP * NPP;     // X = fft2(xp) (row-major)
  float* XI   = ws + o; o += (size_t)MPP * NPP;
  float* AR   = ws + o; o += (size_t)NFP * MPP;     // inv DFT rows, /422
  float* AI   = ws + o; o += (size_t)NFP * MPP;
  float* BTR  = ws + o; o += (size_t)NTP * NPP;     // inv DFT cols, n-major
  float* BTI  = ws + o; o += (size_t)NTP * NPP;
  float* UT   = ws + o; o += (size_t)FCHUNK * NTP * MPP;   // n-major
  float* VT   = ws + o; o += (size_t)FCHUNK * NTP * MPP;
  float* PART = ws + o; o += (size_t)NFILT * NTILED;
  (void)ws_size; (void)in_sizes; (void)n_in; (void)out_size;

  // 1) pad + rescale input, stored time-major (fwd1 B-operand).
  build_xp_kernel<<<dim3((MPP + 255) / 256, NPP), 256, 0, stream>>>(x, XPT);

  // 2) twiddles (forward sign -1; inverse sign +1 with 1/N), zero-padded.
  fill_twiddle_kernel<<<dim3((MPP + 255) / 256, MPP), 256, 0, stream>>>(
      F4R, F4I, MPP, MP, MP, MP, -1.f, 1.f);
  fill_twiddle_kernel<<<dim3((NPP + 255) / 256, NPP), 256, 0, stream>>>(
      GR, GI, NPP, NP, NP, NP, -1.f, 1.f);
  fill_twiddle_kernel<<<dim3((MPP + 255) / 256, NFP), 256, 0, stream>>>(
      AR, AI, MPP, NF_, MP, MP, 1.f, 1.f / (float)MP);
  // B transposed: rows = time crop (390 valid of NTP), cols = l (1170 of NPP).
  fill_twiddle_kernel<<<dim3((NPP + 255) / 256, NTP), 256, 0, stream>>>(
      BTR, BTI, NPP, NT_, NP, NP, 1.f, 1.f / (float)NP);

  // 3) forward FFT2 as two WMMA GEMMs (dual-N, 1184/32 = 37 x-tiles).
  gemm_fwd1_kernel<<<dim3(NPP / 32, MPP / 16), 32, 0, stream>>>(
      F4R, F4I, XPT, TR, TI);
  gemm_fwd2_kernel<<<dim3(NPP / 32, MPP / 16), 32, 0, stream>>>(
      TR, TI, GR, GI, XR, XI);

  // 4) per-filter chunks: fused spectrum-multiply GEMM + projection/reduce.
  for (int f0 = 0; f0 < NFILT; f0 += FCHUNK) {
    int nf = (NFILT - f0) < FCHUNK ? (NFILT - f0) : FCHUNK;
    stageC_kernel<<<dim3(TN4, MPP / 16, nf), 32, 0, stream>>>(
        Hr + (size_t)f0 * FILT_STRIDE, Hi + (size_t)f0 * FILT_STRIDE,
        XR, XI, BTR, BTI, UT, VT);
    stageD_kernel<<<dim3(TN4, TM_D, nf), 32, 0, stream>>>(
        UT, VT, AR, AI, PART + (size_t)f0 * NTILED);
  }

  // 5) final per-filter mean.
  reduce_out_kernel<<<NFILT, 128, 0, stream>>>(PART, out);
}